// RelationNetwork_81784767250498
// MI455X (gfx1250) — compile-verified
//
#include <hip/hip_runtime.h>
#include <math.h>

// ---------------------------------------------------------------------------
// CDNA5 (gfx1250) wave32 WMMA f32 16x16x4 fragments.
// A (16x4, MxK): lane m = lane&15 ; VGPR v holds K = v + 2*(lane>>4)
// B (4x16, KxN): lane n = lane&15 ; VGPR v holds K = v + 2*(lane>>4)
// C/D (16x16)  : lane n = lane&15 ; VGPR r holds M = r + 8*(lane>>4)
// ---------------------------------------------------------------------------
typedef float v2f __attribute__((ext_vector_type(2)));
typedef float v8f __attribute__((ext_vector_type(8)));

__device__ __forceinline__ v8f wmma4(v2f a, v2f b, v8f c) {
  return __builtin_amdgcn_wmma_f32_16x16x4_f32(false, a, false, b, (short)0, c,
                                               false, false);
}

__device__ __forceinline__ float sigf(float x) { return 1.0f / (1.0f + expf(-x)); }

// ---------------------------------------------------------------------------
// Embedding gathers
// ---------------------------------------------------------------------------
__global__ void __launch_bounds__(256)
gather_q_kernel(const int* __restrict__ q, const float* __restrict__ emb,
                float* __restrict__ qe) {
  // qe[t][b][e] = emb[q[b][t]][e]; 32*32*32 elems, exact grid
  const int idx = blockIdx.x * 256 + threadIdx.x;
  const int e = idx & 31;
  const int b = (idx >> 5) & 31;
  const int t = idx >> 10;
  qe[idx] = emb[(size_t)q[b * 32 + t] * 32 + e];
}

__global__ void __launch_bounds__(256)
gather_sf_kernel(const int* __restrict__ sfi, const float* __restrict__ emb,
                 float* __restrict__ se) {
  // se[t][s*32+b][e] = emb[sf[s][b][t]][e]; 64*256*32 elems, exact grid
  const int idx = blockIdx.x * 256 + threadIdx.x;
  const int e = idx & 31;
  const int nrow = (idx >> 5) & 255;   // s*32+b
  const int t = idx >> 13;
  se[idx] = emb[(size_t)sfi[nrow * 64 + t] * 32 + e];
}

// ---------------------------------------------------------------------------
// LSTM: each workgroup owns 32 independent batch rows; h,c,weights in LDS.
// gates (32 x 128) per step = 16 waves x one 16x16 WMMA tile, K=32(x)+32(h).
// ---------------------------------------------------------------------------
__global__ void __launch_bounds__(512)
lstm_kernel(const float* __restrict__ x,    // (T, Ntot, 32)
            const float* __restrict__ Wih,  // (128, 32)
            const float* __restrict__ Whh,  // (128, 32)
            const float* __restrict__ bih, const float* __restrict__ bhh,
            float* __restrict__ hout,       // (Ntot, 32)
            int T, int Ntot) {
  __shared__ float sWih[128][33];
  __shared__ float sWhh[128][33];
  __shared__ float sBias[128];
  __shared__ float sH[32][33];
  __shared__ float sC[32][33];
  __shared__ float sX[32][33];
  __shared__ float sG[32][132];

  const int tid = threadIdx.x;
  const int n0 = blockIdx.x * 32;

  for (int i = tid; i < 128 * 32; i += 512) {
    const int r = i >> 5, c = i & 31;
    sWih[r][c] = Wih[i];
    sWhh[r][c] = Whh[i];
  }
  if (tid < 128) sBias[tid] = bih[tid] + bhh[tid];
  for (int i = tid; i < 32 * 32; i += 512) {
    sH[i >> 5][i & 31] = 0.0f;
    sC[i >> 5][i & 31] = 0.0f;
  }
  __syncthreads();

  const int lane = tid & 31;
  const int wave = tid >> 5;   // 0..15
  const int mt = wave >> 3;    // M tile 0..1
  const int nt = wave & 7;     // N tile 0..7
  const int lm = lane & 15;
  const int lg = lane >> 4;
  const int n = nt * 16 + lm;  // gate column

  for (int t = 0; t < T; ++t) {
    const float* xt = x + ((size_t)t * Ntot + n0) * 32;
    for (int i = tid; i < 32 * 32; i += 512) sX[i >> 5][i & 31] = xt[i];
    __syncthreads();

    v8f acc;
    const float bv = sBias[n];
#pragma unroll
    for (int r = 0; r < 8; ++r) acc[r] = bv;

#pragma unroll
    for (int k0 = 0; k0 < 32; k0 += 4) {
      const int kk = k0 + 2 * lg;
      v2f a, b;
      a[0] = sX[mt * 16 + lm][kk]; a[1] = sX[mt * 16 + lm][kk + 1];
      b[0] = sWih[n][kk];          b[1] = sWih[n][kk + 1];
      acc = wmma4(a, b, acc);
    }
#pragma unroll
    for (int k0 = 0; k0 < 32; k0 += 4) {
      const int kk = k0 + 2 * lg;
      v2f a, b;
      a[0] = sH[mt * 16 + lm][kk]; a[1] = sH[mt * 16 + lm][kk + 1];
      b[0] = sWhh[n][kk];          b[1] = sWhh[n][kk + 1];
      acc = wmma4(a, b, acc);
    }
#pragma unroll
    for (int r = 0; r < 8; ++r) sG[mt * 16 + 8 * lg + r][n] = acc[r];
    __syncthreads();

    // cell update: torch gate order i,f,g,o in columns [0:32][32:64][64:96][96:128]
    for (int i = tid; i < 1024; i += 512) {
      const int b_ = i >> 5;
      const int h_ = i & 31;
      const float gi = sG[b_][h_];
      const float gf = sG[b_][32 + h_];
      const float gg = sG[b_][64 + h_];
      const float go = sG[b_][96 + h_];
      const float cn = sigf(gf) * sC[b_][h_] + sigf(gi) * tanhf(gg);
      sC[b_][h_] = cn;
      sH[b_][h_] = sigf(go) * tanhf(cn);
    }
    __syncthreads();
  }

  for (int i = tid; i < 1024; i += 512)
    hout[(size_t)n0 * 32 + i] = sH[i >> 5][i & 31];
}

// ---------------------------------------------------------------------------
// One 256-wide ReLU layer: out(32x256) = relu(in(32xK) @ W^T + b)
// 16 waves, wave w = N tile; each wave does both M tiles, reusing its B frag.
// ---------------------------------------------------------------------------
__device__ __forceinline__ void rn_layer(const float (*in)[260], float (*out)[260],
                                         const float* __restrict__ W,
                                         const float* __restrict__ bias, int K) {
  const int tid = threadIdx.x;
  const int lane = tid & 31;
  const int nt = tid >> 5;
  const int lm = lane & 15;
  const int lg = lane >> 4;
  const int n = nt * 16 + lm;
  const float bv = bias[n];
  v8f acc0, acc1;
#pragma unroll
  for (int r = 0; r < 8; ++r) { acc0[r] = bv; acc1[r] = bv; }
  const float* Wn = W + (size_t)n * K;
  for (int k0 = 0; k0 < K; k0 += 4) {
    const int kk = k0 + 2 * lg;
    const v2f b = *(const v2f*)(Wn + kk);
    const v2f a0 = *(const v2f*)(&in[lm][kk]);
    const v2f a1 = *(const v2f*)(&in[16 + lm][kk]);
    acc0 = wmma4(a0, b, acc0);
    acc1 = wmma4(a1, b, acc1);
  }
#pragma unroll
  for (int r = 0; r < 8; ++r) {
    float v0 = acc0[r]; v0 = v0 > 0.f ? v0 : 0.f;
    float v1 = acc1[r]; v1 = v1 > 0.f ? v1 : 0.f;
    out[8 * lg + r][n] = v0;
    out[16 + 8 * lg + r][n] = v1;
  }
  __syncthreads();
}

// ---------------------------------------------------------------------------
// Relation g-MLP: one workgroup per ordered pair (i,j); 4 layers in LDS,
// per-pair partial written out (deterministic reduce follows).
// ---------------------------------------------------------------------------
__global__ void __launch_bounds__(512)
relation_kernel(const float* __restrict__ hsf, const float* __restrict__ hq,
                const float* __restrict__ g1_w, const float* __restrict__ g1_b,
                const float* __restrict__ g2_w, const float* __restrict__ g2_b,
                const float* __restrict__ g3_w, const float* __restrict__ g3_b,
                const float* __restrict__ g4_w, const float* __restrict__ g4_b,
                float* __restrict__ partial)  // (64, 32, 256)
{
  __shared__ float bufA[32][260];
  __shared__ float bufB[32][260];
  const int tid = threadIdx.x;
  const int p = blockIdx.x;
  const int i = p >> 3, j = p & 7;

  for (int idx = tid; idx < 32 * 96; idx += 512) {
    const int b = idx / 96;
    const int k = idx - b * 96;
    float v;
    if (k < 32)      v = hsf[(i * 32 + b) * 32 + k];
    else if (k < 64) v = hsf[(j * 32 + b) * 32 + (k - 32)];
    else             v = hq[b * 32 + (k - 64)];
    bufA[b][k] = v;
  }
  __syncthreads();

  rn_layer(bufA, bufB, g1_w, g1_b, 96);
  rn_layer(bufB, bufA, g2_w, g2_b, 256);
  rn_layer(bufA, bufB, g3_w, g3_b, 256);
  rn_layer(bufB, bufA, g4_w, g4_b, 256);

  float* dst = partial + (size_t)p * 8192;
  for (int idx = tid; idx < 8192; idx += 512)
    dst[idx] = bufA[idx >> 8][idx & 255];
}

__global__ void __launch_bounds__(256)
reduce_kernel(const float* __restrict__ partial, float* __restrict__ g_o) {
  const int e = blockIdx.x * 256 + threadIdx.x;  // 8192 elems
  float s = 0.f;
  for (int p = 0; p < 64; ++p) s += partial[(size_t)p * 8192 + e];
  g_o[e] = s;
}

// ---------------------------------------------------------------------------
// f-MLP head layers 1-2: h1 = relu(g_o@f1^T+b1), h2 = relu(h1@f2^T+b2)
// ---------------------------------------------------------------------------
__global__ void __launch_bounds__(512)
fmlp_kernel(const float* __restrict__ g_o, const float* __restrict__ f1_w,
            const float* __restrict__ f1_b, const float* __restrict__ f2_w,
            const float* __restrict__ f2_b, float* __restrict__ h2) {
  __shared__ float sA[32][260];
  __shared__ float sB[32][260];
  const int tid = threadIdx.x;
  for (int i = tid; i < 32 * 256; i += 512) sA[i >> 8][i & 255] = g_o[i];
  __syncthreads();
  rn_layer(sA, sB, f1_w, f1_b, 256);

  const int lane = tid & 31, wave = tid >> 5, lm = lane & 15, lg = lane >> 4;
  for (int half = 0; half < 2; ++half) {   // 512 output cols = 2 x 16 N tiles
    const int n = (wave + 16 * half) * 16 + lm;
    const float bv = f2_b[n];
    v8f acc0, acc1;
#pragma unroll
    for (int r = 0; r < 8; ++r) { acc0[r] = bv; acc1[r] = bv; }
    const float* Wn = f2_w + (size_t)n * 256;
    for (int k0 = 0; k0 < 256; k0 += 4) {
      const int kk = k0 + 2 * lg;
      const v2f b = *(const v2f*)(Wn + kk);
      const v2f a0 = *(const v2f*)(&sB[lm][kk]);
      const v2f a1 = *(const v2f*)(&sB[16 + lm][kk]);
      acc0 = wmma4(a0, b, acc0);
      acc1 = wmma4(a1, b, acc1);
    }
#pragma unroll
    for (int r = 0; r < 8; ++r) {
      float v0 = acc0[r]; v0 = v0 > 0.f ? v0 : 0.f;
      float v1 = acc1[r]; v1 = v1 > 0.f ? v1 : 0.f;
      h2[(8 * lg + r) * 512 + n] = v0;
      h2[(16 + 8 * lg + r) * 512 + n] = v1;
    }
  }
}

// ---------------------------------------------------------------------------
// Final GEMM (bandwidth-dominant): out(32,128000) = h2(32,512)@f3_w^T + f3_b
// 500 blocks x 256 vocab cols. h2 staged in LDS (L2-hot across blocks);
// f3_w (262 MB, single-use, > L2) streamed with NONTEMPORAL loads so it does
// not sweep the 192 MB L2; output (write-once) stored nontemporal as well.
// Each f3_w element is read exactly once via aligned 8B lane loads.
// ---------------------------------------------------------------------------
__global__ void __launch_bounds__(512)
out_gemm_kernel(const float* __restrict__ h2, const float* __restrict__ f3_w,
                const float* __restrict__ f3_b, float* __restrict__ out) {
  __shared__ float sX[32][516];
  const int tid = threadIdx.x;
  for (int i = tid; i < 32 * 512; i += 512) sX[i >> 9][i & 511] = h2[i];
  __syncthreads();

  const int lane = tid & 31, wave = tid >> 5, lm = lane & 15, lg = lane >> 4;
  const int n = blockIdx.x * 256 + wave * 16 + lm;
  const float* Wn = f3_w + (size_t)n * 512;
  const float bv = f3_b[n];
  v8f acc0, acc1;
#pragma unroll
  for (int r = 0; r < 8; ++r) { acc0[r] = bv; acc1[r] = bv; }

  for (int k0 = 0; k0 < 512; k0 += 4) {
    const int kk = k0 + 2 * lg;
    const v2f b = __builtin_nontemporal_load((const v2f*)(Wn + kk));
    const v2f a0 = *(const v2f*)(&sX[lm][kk]);
    const v2f a1 = *(const v2f*)(&sX[16 + lm][kk]);
    acc0 = wmma4(a0, b, acc0);
    acc1 = wmma4(a1, b, acc1);
  }
#pragma unroll
  for (int r = 0; r < 8; ++r) {
    __builtin_nontemporal_store(acc0[r],
        out + (size_t)(8 * lg + r) * 128000 + n);
    __builtin_nontemporal_store(acc1[r],
        out + (size_t)(16 + 8 * lg + r) * 128000 + n);
  }
}

// ---------------------------------------------------------------------------
// Launch. Workspace layout (floats): qe 32768 | se 524288 | hq 1024 |
// hsf 8192 | partial 524288 | g_o 8192 | h2 16384  => ~4.5 MB total.
// ---------------------------------------------------------------------------
extern "C" void kernel_launch(void* const* d_in, const int* in_sizes, int n_in,
                              void* d_out, int out_size, void* d_ws,
                              size_t ws_size, hipStream_t stream) {
  const int* q = (const int*)d_in[0];
  const int* sf = (const int*)d_in[1];
  const float* q_emb = (const float*)d_in[2];
  const float* sf_emb = (const float*)d_in[3];
  const float* q_Wih = (const float*)d_in[4];
  const float* q_Whh = (const float*)d_in[5];
  const float* q_bih = (const float*)d_in[6];
  const float* q_bhh = (const float*)d_in[7];
  const float* sf_Wih = (const float*)d_in[8];
  const float* sf_Whh = (const float*)d_in[9];
  const float* sf_bih = (const float*)d_in[10];
  const float* sf_bhh = (const float*)d_in[11];
  const float* g1_w = (const float*)d_in[12];
  const float* g1_b = (const float*)d_in[13];
  const float* g2_w = (const float*)d_in[14];
  const float* g2_b = (const float*)d_in[15];
  const float* g3_w = (const float*)d_in[16];
  const float* g3_b = (const float*)d_in[17];
  const float* g4_w = (const float*)d_in[18];
  const float* g4_b = (const float*)d_in[19];
  const float* f1_w = (const float*)d_in[20];
  const float* f1_b = (const float*)d_in[21];
  const float* f2_w = (const float*)d_in[22];
  const float* f2_b = (const float*)d_in[23];
  const float* f3_w = (const float*)d_in[24];
  const float* f3_b = (const float*)d_in[25];

  float* ws = (float*)d_ws;
  float* qe = ws;                  // 32768
  float* se = ws + 32768;          // 524288
  float* hq = ws + 557056;         // 1024
  float* hsf = ws + 558080;        // 8192
  float* part = ws + 566272;       // 524288
  float* g_o = ws + 1090560;       // 8192
  float* h2 = ws + 1098752;        // 16384
  float* out = (float*)d_out;      // 32 * 128000

  gather_q_kernel<<<128, 256, 0, stream>>>(q, q_emb, qe);
  gather_sf_kernel<<<2048, 256, 0, stream>>>(sf, sf_emb, se);
  lstm_kernel<<<1, 512, 0, stream>>>(qe, q_Wih, q_Whh, q_bih, q_bhh, hq, 32, 32);
  lstm_kernel<<<8, 512, 0, stream>>>(se, sf_Wih, sf_Whh, sf_bih, sf_bhh, hsf, 64, 256);
  relation_kernel<<<64, 512, 0, stream>>>(hsf, hq, g1_w, g1_b, g2_w, g2_b, g3_w,
                                          g3_b, g4_w, g4_b, part);
  reduce_kernel<<<32, 256, 0, stream>>>(part, g_o);
  fmlp_kernel<<<1, 512, 0, stream>>>(g_o, f1_w, f1_b, f2_w, f2_b, h2);
  out_gemm_kernel<<<500, 512, 0, stream>>>(h2, f3_w, f3_b, out);
}